// ValueModel_57913339019861
// MI455X (gfx1250) — compile-verified
//
#include <hip/hip_runtime.h>
#include <hip/hip_bf16.h>

// Problem constants (match reference)
#define BB 128
#define SS 512
#define DD 1024
#define HH 400
#define LL 20
#define EE 80
#define NE (EE + LL)   // edges + self loops
#define NT 5           // N-tiles (16 cols each) per wave in the WMMA GEMM

typedef float v2f __attribute__((ext_vector_type(2)));
typedef float v8f __attribute__((ext_vector_type(8)));

// ---------------------------------------------------------------------------
// 1) Span sums: x0[b*L+l, :] = sum_{r=start..end} hidden[b, r, :]
//    One block per (b,l); 256 threads * float4 = 1024 cols, fully coalesced.
//    This is the roofline-dominant pass (268 MB of HBM reads); prefetch two
//    rows ahead to keep the load pipe full.
// ---------------------------------------------------------------------------
__global__ __launch_bounds__(256) void spansum_kernel(
    const float* __restrict__ hs, const int* __restrict__ spans,
    float* __restrict__ x0) {
  const int blk = blockIdx.x;            // b*L + l
  const int b   = blk / LL;
  const int st  = spans[blk * 2 + 0];
  const int en  = spans[blk * 2 + 1];
  const int c   = threadIdx.x * 4;
  const float* base = hs + (size_t)b * SS * DD + c;
  float4 acc = make_float4(0.f, 0.f, 0.f, 0.f);
  for (int r = st; r <= en; ++r) {
    if (r + 2 <= en)
      __builtin_prefetch(base + (size_t)(r + 2) * DD, 0, 1);  // global_prefetch_b8
    const float4 v = *(const float4*)(base + (size_t)r * DD);
    acc.x += v.x; acc.y += v.y; acc.z += v.z; acc.w += v.w;
  }
  *(float4*)(x0 + (size_t)blk * DD + c) = acc;
}

// ---------------------------------------------------------------------------
// 2) GCN symmetric norms over E real edges + L self loops (deg from dst).
// ---------------------------------------------------------------------------
__global__ void norm_kernel(const int* __restrict__ edge_index,
                            float* __restrict__ norm) {
  if (threadIdx.x != 0 || blockIdx.x != 0) return;
  float deg[LL];
  for (int l = 0; l < LL; ++l) deg[l] = 1.f;                    // self loop
  for (int e = 0; e < EE; ++e) deg[edge_index[EE + e]] += 1.f;  // dst row
  float dinv[LL];
  for (int l = 0; l < LL; ++l) dinv[l] = (deg[l] > 0.f) ? rsqrtf(deg[l]) : 0.f;
  for (int e = 0; e < EE; ++e)
    norm[e] = dinv[edge_index[e]] * dinv[edge_index[EE + e]];
  for (int l = 0; l < LL; ++l) norm[EE + l] = dinv[l] * dinv[l];
}

// ---------------------------------------------------------------------------
// 3) f32 WMMA GEMM: C[M,N] = A[M,K] @ B[K,N]   (no bias; bias added in prop)
//    Each wave owns one 16-row M-tile and NT=5 16-col N-tiles (16x80 of C):
//      - 5 independent accumulator chains -> 5-way ILP on the matrix pipe
//      - the A fragment load (one b64 per lane) amortized over 5 WMMAs
//    Operand layouts per ISA 7.12.2:
//      A 16x4:  lane l: M = l&15, K-half = (l>>4)*2, VGPR pair = K, K+1
//      B 4x16:  lane l: N = l&15, K-half = (l>>4)*2, VGPR pair = K, K+1
//      C 16x16: VGPR r: lanes 0-15 M=r, lanes 16-31 M=r+8
//    Requires M%16==0, N%(16*NT)==0, K%4==0 (2560, 400, 1024/400: all OK).
// ---------------------------------------------------------------------------
__global__ __launch_bounds__(256) void gemm_f32_wmma_kernel(
    const float* __restrict__ A, const float* __restrict__ Bm,
    float* __restrict__ C, int M, int N, int K) {
  const int wave = (blockIdx.x * blockDim.x + threadIdx.x) >> 5;
  const int lane = threadIdx.x & 31;
  const int ngrp = N / (16 * NT);           // N-groups (400/80 = 5)
  const int ntot = (M >> 4) * ngrp;
  if (wave >= ntot) return;                 // whole-wave exit: EXEC stays all-1s
  const int mt = wave / ngrp, ng = wave % ngrp;
  const int m0 = mt << 4,     n0 = ng * 16 * NT;

  const int mn    = lane & 15;              // row (A) / col (B) within tile
  const int khalf = (lane >> 4) << 1;       // 0 or 2

  const float* aptr = A  + (size_t)(m0 + mn) * K + khalf;   // +k -> A[m,k+khalf..+1]
  const float* bptr = Bm + (size_t)khalf * N + n0 + mn;     // +k*N -> B[k+khalf, n0+mn]

  v8f c[NT];
#pragma unroll
  for (int t = 0; t < NT; ++t) c[t] = (v8f){0.f,0.f,0.f,0.f,0.f,0.f,0.f,0.f};

#pragma unroll 2
  for (int k = 0; k < K; k += 4) {
    __builtin_prefetch(aptr + 16, 0, 1);                 // ~4 k-steps ahead
    __builtin_prefetch(bptr + (size_t)16 * N, 0, 1);
    const v2f a = *(const v2f*)aptr;                     // contiguous pair, 8B aligned
#pragma unroll
    for (int t = 0; t < NT; ++t) {
      v2f b;
      b.x = bptr[16 * t];
      b.y = bptr[16 * t + N];
      c[t] = __builtin_amdgcn_wmma_f32_16x16x4_f32(
          /*neg_a=*/false, a, /*neg_b=*/false, b,
          /*c_mod=*/(short)0, c[t], /*reuse_a=*/false, /*reuse_b=*/false);
    }
    aptr += 4;
    bptr += (size_t)4 * N;
  }

  const int crow = m0 + ((lane >> 4) << 3); // +0 or +8
#pragma unroll
  for (int t = 0; t < NT; ++t) {
    const int ccol = n0 + 16 * t + mn;
#pragma unroll
    for (int r = 0; r < 8; ++r)
      C[(size_t)(crow + r) * N + ccol] = c[t][r];
  }
}

// ---------------------------------------------------------------------------
// 4) Graph propagation + bias + ReLU, per batch sample.
//    out[b,n,h] = relu( bias[h] + sum_{e: dst==n} norm[e] * z[b,src_e,h] )
//    Each thread owns feature columns h = tid, tid+256 -> race-free.
// ---------------------------------------------------------------------------
__global__ __launch_bounds__(256) void prop_kernel(
    const float* __restrict__ z, const int* __restrict__ edge_index,
    const float* __restrict__ norm, const float* __restrict__ bias,
    float* __restrict__ out) {
  __shared__ float zs[LL * HH];             // 32000 B
  __shared__ int   se[NE];
  __shared__ int   de[NE];
  __shared__ float nw[NE];
  const int b   = blockIdx.x;
  const int tid = threadIdx.x;
  const float* zb = z + (size_t)b * LL * HH;
  for (int i = tid; i < LL * HH; i += 256) zs[i] = zb[i];
  if (tid < NE) {
    if (tid < EE) { se[tid] = edge_index[tid]; de[tid] = edge_index[EE + tid]; }
    else          { se[tid] = tid - EE;        de[tid] = tid - EE; }
    nw[tid] = norm[tid];
  }
  __syncthreads();

  float* ob = out + (size_t)b * LL * HH;
  for (int h = tid; h < HH; h += 256) {
    const float bh = bias[h];
    for (int n = 0; n < LL; ++n) {
      float a = bh;
      for (int e = 0; e < NE; ++e)
        if (de[e] == n) a += nw[e] * zs[se[e] * HH + h];
      ob[n * HH + h] = a > 0.f ? a : 0.f;
    }
  }
}

// ---------------------------------------------------------------------------
// 5) Per-label heads: out_logits[b,l,k] = sum_h x[b,l,h]*head_W[l,h,k] + head_b[l,k]
// ---------------------------------------------------------------------------
__global__ __launch_bounds__(128) void head_kernel(
    const float* __restrict__ x, const float* __restrict__ hW,
    const float* __restrict__ hb, float* __restrict__ out) {
  const int blk = blockIdx.x;               // b*L + l
  const int l   = blk % LL;
  const int tid = threadIdx.x;
  const float* xr = x  + (size_t)blk * HH;
  const float* w  = hW + (size_t)l * HH * 2;
  float a0 = 0.f, a1 = 0.f;
  for (int h = tid; h < HH; h += 128) {
    const float xv = xr[h];
    a0 += xv * w[h * 2 + 0];
    a1 += xv * w[h * 2 + 1];
  }
  __shared__ float r0[128];
  __shared__ float r1[128];
  r0[tid] = a0; r1[tid] = a1;
  __syncthreads();
  for (int s = 64; s > 0; s >>= 1) {
    if (tid < s) { r0[tid] += r0[tid + s]; r1[tid] += r1[tid + s]; }
    __syncthreads();
  }
  if (tid == 0) {
    out[blk * 2 + 0] = r0[0] + hb[l * 2 + 0];
    out[blk * 2 + 1] = r1[0] + hb[l * 2 + 1];
  }
}

// ---------------------------------------------------------------------------
// Launch: spansum -> norms -> [gemm -> prop] x3 -> head
// ---------------------------------------------------------------------------
extern "C" void kernel_launch(void* const* d_in, const int* in_sizes, int n_in,
                              void* d_out, int out_size, void* d_ws, size_t ws_size,
                              hipStream_t stream) {
  const float* hidden = (const float*)d_in[0];
  const int*   spans  = (const int*)  d_in[1];
  const int*   eidx   = (const int*)  d_in[2];
  // d_in[3] = edge_attr (unused, matches reference)
  const float* W1 = (const float*)d_in[4];
  const float* b1 = (const float*)d_in[5];
  const float* W2 = (const float*)d_in[6];
  const float* b2 = (const float*)d_in[7];
  const float* W3 = (const float*)d_in[8];
  const float* b3 = (const float*)d_in[9];
  const float* hW = (const float*)d_in[10];
  const float* hb = (const float*)d_in[11];

  float* out_logits = (float*)d_out;                       // [B,L,2]  = 5120
  float* x_final    = (float*)d_out + (size_t)BB * LL * 2; // [B,L*H]  = 1,024,000

  // Workspace layout
  char* ws = (char*)d_ws;
  float* x0   = (float*)ws;                                  ws += (size_t)BB * LL * DD * 4; // 10.49 MB
  float* zbuf = (float*)ws;                                  ws += (size_t)BB * LL * HH * 4; //  4.10 MB
  float* xA   = (float*)ws;                                  ws += (size_t)BB * LL * HH * 4; //  4.10 MB
  float* nrm  = (float*)ws;                                  // NE floats

  const int M = BB * LL;                    // 2560 rows
  const int waves = (M >> 4) * (HH / (16 * NT));   // 160 * 5 = 800 waves
  const int gblocks = (waves + 7) / 8;             // 8 waves / 256-thread block

  // 1) span sums
  spansum_kernel<<<BB * LL, 256, 0, stream>>>(hidden, spans, x0);
  // 2) edge norms
  norm_kernel<<<1, 32, 0, stream>>>(eidx, nrm);

  // Layer 1: z = x0 @ W1 ; xA = relu(prop(z) + b1)
  gemm_f32_wmma_kernel<<<gblocks, 256, 0, stream>>>(x0, W1, zbuf, M, HH, DD);
  prop_kernel<<<BB, 256, 0, stream>>>(zbuf, eidx, nrm, b1, xA);

  // Layer 2: z = xA @ W2 ; xA = relu(prop(z) + b2)
  gemm_f32_wmma_kernel<<<gblocks, 256, 0, stream>>>(xA, W2, zbuf, M, HH, HH);
  prop_kernel<<<BB, 256, 0, stream>>>(zbuf, eidx, nrm, b2, xA);

  // Layer 3: z = xA @ W3 ; x_final = relu(prop(z) + b3)   (written into d_out)
  gemm_f32_wmma_kernel<<<gblocks, 256, 0, stream>>>(xA, W3, zbuf, M, HH, HH);
  prop_kernel<<<BB, 256, 0, stream>>>(zbuf, eidx, nrm, b3, x_final);

  // Heads
  head_kernel<<<BB * LL, 128, 0, stream>>>(x_final, hW, hb, out_logits);
}